// EmotionGATv2_33904471835079
// MI455X (gfx1250) — compile-verified
//
#include <hip/hip_runtime.h>
#include <math.h>

typedef __attribute__((ext_vector_type(2))) float v2f;
typedef __attribute__((ext_vector_type(8))) float v8f;

#define WMMA_F32(a, b, c) \
  __builtin_amdgcn_wmma_f32_16x16x4_f32(false, (a), false, (b), (short)0, (c), false, false)

// ---------- helpers ----------
__device__ __forceinline__ void atomicMaxF(float* addr, float val) {
  // IEEE-754 ordering trick: works for mixed signs, -inf init
  if (val >= 0.0f) atomicMax((int*)addr, __float_as_int(val));
  else             atomicMin((unsigned int*)addr, (unsigned int)__float_as_int(val));
}

// ---------- WMMA fp32 GEMM ----------
// DUAL=true : each wave computes one 16x16 tile of C1=A@B1 and the same tile of C2=A@B2
//             (A-fragment reused across the two WMMA chains).
// DUAL=false: each wave computes a 16x32 strip of C1=A@B1 (two adjacent column tiles,
//             A-fragment reused). Requires (Nc/16) even.
// K must be a multiple of 8. Two k-steps of loads are issued before the 4 WMMAs so the
// memory clauses overlap; B pointers advance by increments (no per-step address rebuild).
template<bool DUAL>
__global__ void wmma_gemm_f32(const float* __restrict__ A,
                              const float* __restrict__ B1, float* __restrict__ C1,
                              const float* __restrict__ B2, float* __restrict__ C2,
                              const float* __restrict__ bias,
                              int M, int Nc, int K)
{
  const int lane   = threadIdx.x & 31;
  const int wv     = threadIdx.x >> 5;
  const int tilesN = Nc >> 4;
  const int tilesM = (M + 15) >> 4;
  const int unitsN = DUAL ? tilesN : (tilesN >> 1);
  const int work   = blockIdx.x * (blockDim.x >> 5) + wv;
  if (work >= tilesM * unitsN) return;   // whole-wave exit: EXEC stays all-ones for WMMA
  const int rt   = work / unitsN;
  const int half = lane >> 4;            // 0: lanes 0-15, 1: lanes 16-31
  const int kh   = half << 1;            // k offset 0 or 2 inside each 4-wide step
  int m = (rt << 4) + (lane & 15); if (m >= M) m = M - 1;   // clamp, no divergence
  const float* Arow = A + (size_t)m * K;

  int n0, n1;
  const float *P0, *P1;
  if (DUAL) {
    const int ct = work % unitsN;
    n0 = (ct << 4) + (lane & 15);
    n1 = n0;
    P0 = B1 + (size_t)kh * Nc + n0;
    P1 = B2 + (size_t)kh * Nc + n1;
  } else {
    const int ctp = work % unitsN;
    n0 = (ctp << 5) + (lane & 15);
    n1 = n0 + 16;
    P0 = B1 + (size_t)kh * Nc + n0;
    P1 = B1 + (size_t)kh * Nc + n1;
  }
  const size_t sB = (size_t)4 * Nc;      // B row-stride per k-step

  v8f acc0 = {0.f,0.f,0.f,0.f,0.f,0.f,0.f,0.f};
  v8f acc1 = acc0;
  for (int kk = 0; kk < K; kk += 8) {
    v2f a0, a1, b00, b01, b10, b11;
    a0.x  = Arow[kk + kh];     a0.y  = Arow[kk + kh + 1];
    a1.x  = Arow[kk + 4 + kh]; a1.y  = Arow[kk + 4 + kh + 1];
    b00.x = P0[0];             b00.y = P0[Nc];
    b01.x = P1[0];             b01.y = P1[Nc];
    b10.x = P0[sB];            b10.y = P0[sB + Nc];
    b11.x = P1[sB];            b11.y = P1[sB + Nc];
    acc0 = WMMA_F32(a0, b00, acc0);
    acc1 = WMMA_F32(a0, b01, acc1);
    acc0 = WMMA_F32(a1, b10, acc0);
    acc1 = WMMA_F32(a1, b11, acc1);
    P0 += 2 * sB;
    P1 += 2 * sB;
  }

  const float bj0 = bias ? bias[n0] : 0.0f;
  const float bj1 = bias ? bias[n1] : 0.0f;
  const int rbase = (rt << 4) + (half << 3);
  for (int r = 0; r < 8; ++r) {
    const int row = rbase + r;
    if (row < M) {
      C1[(size_t)row * Nc + n0] = acc0[r] + bj0;
      if (DUAL) C2[(size_t)row * Nc + n1] = acc1[r];
      else      C1[(size_t)row * Nc + n1] = acc1[r] + bj1;
    }
  }
}

// ---------- edge pass 1: alpha[e,h] = att_h . leaky(xl[src]+xr[dst]); segment max ----------
template<int C>
__global__ void edge_alpha_kernel(const float* __restrict__ xl, const float* __restrict__ xr,
                                  const float* __restrict__ att, const int* __restrict__ ei,
                                  float* __restrict__ alpha, float* __restrict__ mx,
                                  int E, int Nn)
{
  const int lane = threadIdx.x & 31;
  const int e = blockIdx.x * (blockDim.x >> 5) + (threadIdx.x >> 5);
  if (e >= E + Nn) return;
  int s, d;
  if (e < E) { s = ei[e]; d = ei[E + e]; } else { s = e - E; d = s; }
  const float* pl = xl + (size_t)s * (4 * C);
  const float* pr = xr + (size_t)d * (4 * C);
  float part[4] = {0.f, 0.f, 0.f, 0.f};
#pragma unroll
  for (int i = 0; i < (4 * C) / 32; ++i) {
    const int j = lane + i * 32;
    float v = pl[j] + pr[j];
    v = v > 0.f ? v : 0.2f * v;              // leaky_relu(0.2)
    part[(i * 32) / C] += v * att[j];        // head index is compile-time per i
  }
#pragma unroll
  for (int h = 0; h < 4; ++h) {
    float v = part[h];
    for (int off = 16; off > 0; off >>= 1) v += __shfl_xor(v, off, 32);
    if (lane == 0) {
      alpha[(size_t)e * 4 + h] = v;
      atomicMaxF(mx + (size_t)d * 4 + h, v);
    }
  }
}

// ---------- edge pass 2: e = exp(alpha - mx[dst]); denom[dst] += e ----------
__global__ void edge_exp_kernel(float* __restrict__ alpha, const float* __restrict__ mx,
                                float* __restrict__ den, const int* __restrict__ ei,
                                int E, int Nn)
{
  const int tid = blockIdx.x * blockDim.x + threadIdx.x;
  if (tid >= (E + Nn) * 4) return;
  const int e = tid >> 2, h = tid & 3;
  const int d = (e < E) ? ei[E + e] : (e - E);
  const float v = __expf(alpha[tid] - mx[d * 4 + h]);
  alpha[tid] = v;
  atomicAdd(den + d * 4 + h, v);
}

// ---------- edge pass 3: out[dst] += xl[src] * (e/denom[dst]) ----------
template<int C>
__global__ void edge_scatter_kernel(const float* __restrict__ xl, const float* __restrict__ eexp,
                                    const float* __restrict__ den, const int* __restrict__ ei,
                                    float* __restrict__ out, int E, int Nn)
{
  const int lane = threadIdx.x & 31;
  const int e = blockIdx.x * (blockDim.x >> 5) + (threadIdx.x >> 5);
  if (e >= E + Nn) return;
  int s, d;
  if (e < E) { s = ei[e]; d = ei[E + e]; } else { s = e - E; d = s; }
  float a[4];
#pragma unroll
  for (int h = 0; h < 4; ++h) a[h] = eexp[(size_t)e * 4 + h] / den[(size_t)d * 4 + h];
  const float* pl = xl + (size_t)s * (4 * C);
  float* po = out + (size_t)d * (4 * C);
#pragma unroll
  for (int i = 0; i < (4 * C) / 32; ++i) {
    const int j = lane + i * 32;
    atomicAdd(po + j, pl[j] * a[(i * 32) / C]);
  }
}

// ---------- BN stats: one block per column ----------
__global__ void bn_stats_kernel(const float* __restrict__ X, int rows, int cols,
                                float* __restrict__ mean, float* __restrict__ rstd)
{
  __shared__ float sh0[256], sh1[256];
  const int c = blockIdx.x, t = threadIdx.x;
  float s = 0.f, s2 = 0.f;
  for (int i = t; i < rows; i += blockDim.x) {
    const float v = X[(size_t)i * cols + c];
    s += v; s2 += v * v;
  }
  sh0[t] = s; sh1[t] = s2;
  __syncthreads();
  for (int o = blockDim.x >> 1; o > 0; o >>= 1) {
    if (t < o) { sh0[t] += sh0[t + o]; sh1[t] += sh1[t + o]; }
    __syncthreads();
  }
  if (t == 0) {
    const float m = sh0[0] / rows;
    const float var = sh1[0] / rows - m * m;   // biased variance (training math)
    mean[c] = m;
    rstd[c] = rsqrtf(var + 1e-5f);
  }
}

__global__ void bn_elu_kernel(float* __restrict__ X, const float* __restrict__ mean,
                              const float* __restrict__ rstd, const float* __restrict__ g,
                              const float* __restrict__ b, size_t total, int cols)
{
  const size_t tid = (size_t)blockIdx.x * blockDim.x + threadIdx.x;
  if (tid >= total) return;
  const int j = (int)(tid % cols);
  const float v = (X[tid] - mean[j]) * rstd[j] * g[j] + b[j];
  X[tid] = v > 0.f ? v : (__expf(v) - 1.f);
}

// x2 = elu(bn(h2) + res + bres), written into res
__global__ void bn_res_elu_kernel(const float* __restrict__ H2, float* __restrict__ R,
                                  const float* __restrict__ mean, const float* __restrict__ rstd,
                                  const float* __restrict__ g, const float* __restrict__ b,
                                  const float* __restrict__ bres, size_t total, int cols)
{
  const size_t tid = (size_t)blockIdx.x * blockDim.x + threadIdx.x;
  if (tid >= total) return;
  const int j = (int)(tid % cols);
  const float v = (H2[tid] - mean[j]) * rstd[j] * g[j] + b[j] + R[tid] + bres[j];
  R[tid] = v > 0.f ? v : (__expf(v) - 1.f);
}

__global__ void fill_kernel(float* __restrict__ p, float v, size_t n) {
  const size_t tid = (size_t)blockIdx.x * blockDim.x + threadIdx.x;
  if (tid < n) p[tid] = v;
}

__global__ void init_bias_kernel(float* __restrict__ out, const float* __restrict__ bias,
                                 size_t total, int cols) {
  const size_t tid = (size_t)blockIdx.x * blockDim.x + threadIdx.x;
  if (tid < total) out[tid] = bias[tid % cols];
}

// ---------- pooling ----------
__global__ void pool_init_kernel(float* __restrict__ pool) {
  const int tid = blockIdx.x * blockDim.x + threadIdx.x;
  if (tid >= 64 * 512) return;
  pool[tid] = ((tid & 511) < 256) ? 0.f : -__builtin_inff();
}

__global__ void pool_scatter_kernel(const float* __restrict__ x2, const int* __restrict__ batch,
                                    float* __restrict__ pool, float* __restrict__ cnt, int Nn)
{
  const int lane = threadIdx.x & 31;
  const int i = blockIdx.x * (blockDim.x >> 5) + (threadIdx.x >> 5);
  if (i >= Nn) return;
  const int g = batch[i];
  if (lane == 0) atomicAdd(cnt + g, 1.0f);
  const float* px = x2 + (size_t)i * 256;
  float* pp = pool + (size_t)g * 512;
#pragma unroll
  for (int t = 0; t < 8; ++t) {
    const int j = lane + 32 * t;
    const float v = px[j];
    atomicAdd(pp + j, v);
    atomicMaxF(pp + 256 + j, v);
  }
}

__global__ void pool_finalize_kernel(float* __restrict__ pool, const float* __restrict__ cnt) {
  const int tid = blockIdx.x * blockDim.x + threadIdx.x;
  if (tid >= 64 * 512) return;
  const int g = tid >> 9, j = tid & 511;
  const float c = cnt[g];
  if (j < 256) pool[tid] = pool[tid] / fmaxf(c, 1.f);
  else         pool[tid] = (c > 0.f) ? pool[tid] : 0.f;
}

// ---------- classifier tail: relu(bn(hc)) @ Wc2 + bc2 ----------
__global__ void classifier_final(const float* __restrict__ hc, const float* __restrict__ mean,
                                 const float* __restrict__ rstd, const float* __restrict__ gc,
                                 const float* __restrict__ bec, const float* __restrict__ Wc2,
                                 const float* __restrict__ bc2, float* __restrict__ out)
{
  __shared__ float r[128];
  const int g = blockIdx.x, t = threadIdx.x;   // 128 threads
  const float v = (hc[(size_t)g * 128 + t] - mean[t]) * rstd[t] * gc[t] + bec[t];
  r[t] = v > 0.f ? v : 0.f;
  __syncthreads();
  if (t < 8) {
    float s = bc2[t];
    for (int j = 0; j < 128; ++j) s += r[j] * Wc2[j * 8 + t];
    out[g * 8 + t] = s;
  }
}

// ---------- launch ----------
extern "C" void kernel_launch(void* const* d_in, const int* in_sizes, int n_in,
                              void* d_out, int out_size, void* d_ws, size_t ws_size,
                              hipStream_t stream) {
  const float* x    = (const float*)d_in[0];
  const float* Wl1  = (const float*)d_in[1];
  const float* Wr1  = (const float*)d_in[2];
  const float* att1 = (const float*)d_in[3];
  const float* b1   = (const float*)d_in[4];
  const float* g1   = (const float*)d_in[5];
  const float* be1  = (const float*)d_in[6];
  const float* Wl2  = (const float*)d_in[7];
  const float* Wr2  = (const float*)d_in[8];
  const float* att2 = (const float*)d_in[9];
  const float* b2   = (const float*)d_in[10];
  const float* g2   = (const float*)d_in[11];
  const float* be2  = (const float*)d_in[12];
  const float* Wres = (const float*)d_in[13];
  const float* bres = (const float*)d_in[14];
  const float* Wc1  = (const float*)d_in[15];
  const float* bc1  = (const float*)d_in[16];
  const float* gc   = (const float*)d_in[17];
  const float* bec  = (const float*)d_in[18];
  const float* Wc2  = (const float*)d_in[19];
  const float* bc2  = (const float*)d_in[20];
  const int*   ei   = (const int*)d_in[21];
  const int*   batch= (const int*)d_in[22];

  const int N = in_sizes[0] / 128;
  const int E = in_sizes[21] / 2;
  const int EN = E + N;
  const int G = 64;

  // workspace layout (floats), with reuse: xl2 overlays xl1+xr1 (dead after conv1 edges)
  float* W = (float*)d_ws;
  float* xl1 = W;                              // N*128
  float* xr1 = W + (size_t)N * 128;            // N*128
  float* h1  = W + (size_t)N * 256;            // N*128 (conv1 out -> x1 in-place)
  float* xl2 = W;                              // N*256 (overlays xl1,xr1)
  float* xr2 = W + (size_t)N * 384;            // N*256
  float* h2  = W + (size_t)N * 640;            // N*256
  float* res = W + (size_t)N * 896;            // N*256 (x1@Wres -> x2 in-place)
  float* alpha = W + (size_t)N * 1152;         // EN*4
  float* mx  = alpha + (size_t)EN * 4;         // N*4
  float* den = mx + (size_t)N * 4;             // N*4
  float* mean = den + (size_t)N * 4;           // 512
  float* rstd = mean + 512;                    // 512
  float* pool = rstd + 512;                    // G*512
  float* cnt  = pool + (size_t)G * 512;        // G
  float* hc   = cnt + G;                       // G*128
  (void)ws_size; (void)n_in; (void)out_size;

  auto cdiv = [](long long a, long long b) { return (int)((a + b - 1) / b); };

  // ===== conv1 =====
  wmma_gemm_f32<true><<<cdiv((long long)(N/16)*(128/16), 8), 256, 0, stream>>>(
      x, Wl1, xl1, Wr1, xr1, nullptr, N, 128, 128);
  fill_kernel<<<cdiv((long long)N*4, 256), 256, 0, stream>>>(mx, -__builtin_inff(), (size_t)N*4);
  fill_kernel<<<cdiv((long long)N*4, 256), 256, 0, stream>>>(den, 0.f, (size_t)N*4);
  init_bias_kernel<<<cdiv((long long)N*128, 256), 256, 0, stream>>>(h1, b1, (size_t)N*128, 128);
  edge_alpha_kernel<32><<<cdiv(EN, 8), 256, 0, stream>>>(xl1, xr1, att1, ei, alpha, mx, E, N);
  edge_exp_kernel<<<cdiv((long long)EN*4, 256), 256, 0, stream>>>(alpha, mx, den, ei, E, N);
  edge_scatter_kernel<32><<<cdiv(EN, 8), 256, 0, stream>>>(xl1, alpha, den, ei, h1, E, N);
  bn_stats_kernel<<<128, 256, 0, stream>>>(h1, N, 128, mean, rstd);
  bn_elu_kernel<<<cdiv((long long)N*128, 256), 256, 0, stream>>>(h1, mean, rstd, g1, be1, (size_t)N*128, 128);

  // ===== conv2 =====
  wmma_gemm_f32<true><<<cdiv((long long)(N/16)*(256/16), 8), 256, 0, stream>>>(
      h1, Wl2, xl2, Wr2, xr2, nullptr, N, 256, 128);
  fill_kernel<<<cdiv((long long)N*4, 256), 256, 0, stream>>>(mx, -__builtin_inff(), (size_t)N*4);
  fill_kernel<<<cdiv((long long)N*4, 256), 256, 0, stream>>>(den, 0.f, (size_t)N*4);
  init_bias_kernel<<<cdiv((long long)N*256, 256), 256, 0, stream>>>(h2, b2, (size_t)N*256, 256);
  edge_alpha_kernel<64><<<cdiv(EN, 8), 256, 0, stream>>>(xl2, xr2, att2, ei, alpha, mx, E, N);
  edge_exp_kernel<<<cdiv((long long)EN*4, 256), 256, 0, stream>>>(alpha, mx, den, ei, E, N);
  edge_scatter_kernel<64><<<cdiv(EN, 8), 256, 0, stream>>>(xl2, alpha, den, ei, h2, E, N);
  bn_stats_kernel<<<256, 256, 0, stream>>>(h2, N, 256, mean, rstd);

  // ===== residual + elu ===== (single-B GEMM: 16x32 strips, (256/16)/2 = 8 pairs/row)
  wmma_gemm_f32<false><<<cdiv((long long)(N/16)*(256/32), 8), 256, 0, stream>>>(
      h1, Wres, res, nullptr, nullptr, nullptr, N, 256, 128);
  bn_res_elu_kernel<<<cdiv((long long)N*256, 256), 256, 0, stream>>>(
      h2, res, mean, rstd, g2, be2, bres, (size_t)N*256, 256);

  // ===== pooling =====
  pool_init_kernel<<<cdiv(64*512, 256), 256, 0, stream>>>(pool);
  fill_kernel<<<1, 64, 0, stream>>>(cnt, 0.f, (size_t)G);
  pool_scatter_kernel<<<cdiv(N, 8), 256, 0, stream>>>(res, batch, pool, cnt, N);
  pool_finalize_kernel<<<cdiv(64*512, 256), 256, 0, stream>>>(pool, cnt);

  // ===== classifier ===== (single-B GEMM: M=64 -> 4 row tiles, 128/32 = 4 pairs/row)
  wmma_gemm_f32<false><<<cdiv((64/16)*(128/32), 8), 256, 0, stream>>>(
      pool, Wc1, hc, nullptr, nullptr, bc1, 64, 128, 512);
  bn_stats_kernel<<<128, 256, 0, stream>>>(hc, 64, 128, mean, rstd);
  classifier_final<<<64, 128, 0, stream>>>(hc, mean, rstd, gc, bec, Wc2, bc2, (float*)d_out);
}